// ChamferLoss_30966714204482
// MI455X (gfx1250) — compile-verified
//
#include <hip/hip_runtime.h>
#include <math.h>

typedef __attribute__((ext_vector_type(2))) float v2f;
typedef __attribute__((ext_vector_type(8))) float v8f;

#define CHUNK 2048  // candidates staged in LDS per pass (32 KB)

// One direction of Chamfer: for each query point, min distance over all
// candidate points. Per-wave 16-query tile; WMMA f32 16x16x4 computes
// (c.c - 2 q.c) for a 16x16 tile per instruction. LDS holds candidates
// pre-swizzled into the exact WMMA B-operand lane layout:
//   tile t, lane l<16 : (x, y)  of candidate t*16+l        (K0,K1)
//   tile t, lane l>=16: (z, c.c) of candidate t*16+(l-16)  (K2,K3)
__global__ __launch_bounds__(256) void chamfer_dir_kernel(
    const float* __restrict__ qpts,   // (B, nq, 3) query points
    const float* __restrict__ cpts,   // (B, nc, 3) candidate points
    int nq, int nc, float weight, float* __restrict__ out)
{
    __shared__ v2f sB[CHUNK * 2];     // 2 lane-slots of 8B per candidate

    const int  tid   = threadIdx.x;
    const int  lane  = tid & 31;
    const int  row   = lane & 15;     // row of the 16-query tile this lane feeds
    const bool hi    = lane >= 16;    // upper half of wave holds K2/K3 operands
    const int  wave  = tid >> 5;
    const int  batch = blockIdx.y;
    const int  qidx  = (blockIdx.x * 8 + wave) * 16 + row;

    const float* qb = qpts + (size_t)batch * nq * 3;
    const float* cb = cpts + (size_t)batch * nc * 3;

    // Augmented A operand: row = (-2qx, -2qy, -2qz, 1)
    //   lanes 0-15: (K0,K1), lanes 16-31: (K2,K3)
    const float qx = qb[3 * qidx + 0];
    const float qy = qb[3 * qidx + 1];
    const float qz = qb[3 * qidx + 2];
    const float xx = qx * qx + qy * qy + qz * qz;   // added after the min

    v2f A;
    A.x = hi ? (-2.0f * qz) : (-2.0f * qx);
    A.y = hi ? 1.0f         : (-2.0f * qy);

    v8f acc;
#pragma unroll
    for (int i = 0; i < 8; ++i) acc[i] = 3.402823466e38f;
    const v8f czero = {};

    for (int base = 0; base < nc; base += CHUNK) {
        // Cooperative stage of CHUNK candidates, swizzled to B-operand layout.
#pragma unroll
        for (int i = 0; i < CHUNK / 256; ++i) {
            const int li = i * 256 + tid;           // candidate within chunk
            const int ci = base + li;
            const float cx = cb[3 * ci + 0];
            const float cy = cb[3 * ci + 1];
            const float cz = cb[3 * ci + 2];
            const float cc = cx * cx + cy * cy + cz * cz;
            const int t = li >> 4;                  // tile within chunk
            const int r = li & 15;                  // column within tile
            v2f lo; lo.x = cx; lo.y = cy;
            v2f hi2; hi2.x = cz; hi2.y = cc;
            sB[(t << 5) + r]      = lo;             // lanes 0-15 slot
            sB[(t << 5) + 16 + r] = hi2;            // lanes 16-31 slot
        }
        __syncthreads();

#pragma unroll 4
        for (int t = 0; t < CHUNK / 16; ++t) {
            const v2f Bv = sB[(t << 5) + lane];     // ds_load_b64, no selects
            // D[n,m] = c.c - 2 q.c  (C = 0)
            v8f d = __builtin_amdgcn_wmma_f32_16x16x4_f32(
                false, A, false, Bv, (short)0, czero, false, false);
#pragma unroll
            for (int i = 0; i < 8; ++i) acc[i] = fminf(acc[i], d[i]);
        }
        __syncthreads();
    }

    // Min across the 16 columns held in each 16-lane half (D layout:
    // VGPR v = rows v / v+8 in the two halves, N = lane%16).
#pragma unroll
    for (int m = 1; m <= 8; m <<= 1) {
#pragma unroll
        for (int i = 0; i < 8; ++i)
            acc[i] = fminf(acc[i], __shfl_xor(acc[i], m, 32));
    }

    // Add per-row q.q, clip, +eps, sqrt, sum the 16 rows of this wave's tile.
    // Row v's xx lives in lane v (rows 0-7) / lane 24+v (rows 8-15).
    float sum = 0.0f;
#pragma unroll
    for (int v = 0; v < 8; ++v) {
        const int src = v + (hi ? 24 : 0);
        const float xr = __shfl(xx, src, 32);
        const float d2 = fmaxf(acc[v] + xr, 0.0f) + 1e-12f;
        sum += sqrtf(d2);
    }
    sum += __shfl_xor(sum, 16, 32);   // combine rows 0-7 and 8-15 halves

    if (lane == 0) atomicAdd(out, weight * sum);
}

__global__ void chamfer_zero_kernel(float* out) {
    if (threadIdx.x == 0) out[0] = 0.0f;
}

extern "C" void kernel_launch(void* const* d_in, const int* in_sizes, int n_in,
                              void* d_out, int out_size, void* d_ws, size_t ws_size,
                              hipStream_t stream) {
    (void)n_in; (void)out_size; (void)d_ws; (void)ws_size;
    const float* target = (const float*)d_in[0];  // (B, M, 3)
    const float* output = (const float*)d_in[1];  // (B, N, 3)
    float* out = (float*)d_out;

    const int B = 4;
    const int n = in_sizes[0] / (3 * B);          // 8192 points per cloud

    dim3 grid(n / 128, B);                        // 8 waves x 16 queries per block

    chamfer_zero_kernel<<<1, 1, 0, stream>>>(out);
    // loss_a_to_b: queries = output, candidates = target; mean over B -> 1/4
    chamfer_dir_kernel<<<grid, 256, 0, stream>>>(output, target, n, n, 0.25f, out);
    // loss_b_to_a weighted 2x (/0.5 in reference): queries = target -> 2/4
    chamfer_dir_kernel<<<grid, 256, 0, stream>>>(target, output, n, n, 0.50f, out);
}